// FrequencyMoEBranch_45372034515270
// MI455X (gfx1250) — compile-verified
//
#include <hip/hip_runtime.h>

typedef float v2f __attribute__((ext_vector_type(2)));
typedef float v8f __attribute__((ext_vector_type(8)));

#define NFFT  256
#define HOP_  64
#define PAD_  128
#define B_    128
#define L_    16384
#define F_    129    // rfft bins
#define T_    257    // frames
#define H_    128    // gate hidden
#define E_    8      // experts
#define C_    64     // conv channels
#define T1_   128    // after conv1 pool2
#define T2_   64     // after conv2 pool2 (== conv3 length)

// fp32 WMMA: D(16x16) += A(16x4) * B(4x16), full wave32 op.
__device__ __forceinline__ v8f wmma4(v2f a, v2f b, v8f c) {
  return __builtin_amdgcn_wmma_f32_16x16x4_f32(false, a, false, b, (short)0, c, false, false);
}

// Low 32 bits of a generic pointer to __shared__ = byte offset in the WG's LDS
// allocation, which is what GLOBAL_LOAD_ASYNC_TO_LDS's VDST expects.
__device__ __forceinline__ uint32_t lds_off(const void* p) {
  return (uint32_t)(uintptr_t)p;
}

// ---------------------------------------------------------------------------
// Kernel 1: fused window + DFT magnitude via WMMA.
// grid (9 f-tiles, 128 b), block 128 (4 waves). Wave handles t-tiles w, w+4, ...
// The Hann window is folded into the LDS cos/sin tables (win depends only on
// the K index), so the A fragment is a raw reflected sample: the global load
// is unconditional (branchless reflect is always in-bounds) and unroll-4 lets
// the scheduler batch 8 loads per wait instead of serializing per k-step.
// ---------------------------------------------------------------------------
__global__ __launch_bounds__(128) void stft_mag_kernel(const float* __restrict__ x,
                                                       float* __restrict__ mag) {
  __shared__ float s_cos[NFFT * 16];   // [n][f_local], win-weighted
  __shared__ float s_sin[NFFT * 16];
  const int fb  = blockIdx.x * 16;
  const int b   = blockIdx.y;
  const int tid = threadIdx.x;

  const float twoPiOverN = 6.28318530717958647692f / (float)NFFT;
  for (int i = tid; i < NFFT * 16; i += 128) {
    const int n = i >> 4, fl = i & 15;
    const int m = ((fb + fl) * n) & (NFFT - 1);   // exact angle reduction mod N
    const float win = 0.5f * (1.0f - cosf(twoPiOverN * (float)n));
    float s, c;
    sincosf(-twoPiOverN * (float)m, &s, &c);
    s_cos[i] = win * c;
    s_sin[i] = win * s;
  }
  __syncthreads();

  const int lane = tid & 31, wave = tid >> 5;
  const int half = lane >> 4, l15 = lane & 15;
  const float* xb = x + (size_t)b * L_;

  for (int tt = wave; tt < 17; tt += 4) {
    const int tbase = tt * 16;
    const int frame = tbase + l15;           // A row for this lane
    const bool fok  = frame < T_;            // only last tile has invalid rows
    const int sbase = frame * HOP_ - PAD_;
    v8f acc_re = {}, acc_im = {};
    #pragma unroll 4
    for (int kk = 0; kk < NFFT; kk += 4) {
      v2f a, bc, bs;
      #pragma unroll
      for (int j = 0; j < 2; ++j) {
        const int n = kk + 2 * half + j;     // K index per A/B fragment layout
        int s = sbase + n;                   // branchless reflect, in-bounds
        s = s < 0 ? -s : s;
        s = min(s, 2 * L_ - 2 - s);
        const float xv = xb[s];
        a[j]  = fok ? xv : 0.0f;
        bc[j] = s_cos[(n << 4) + l15];
        bs[j] = s_sin[(n << 4) + l15];
      }
      acc_re = wmma4(a, bc, acc_re);
      acc_im = wmma4(a, bs, acc_im);
    }
    const int f = fb + l15;                  // D: lane -> freq column
    if (f < F_) {
      #pragma unroll
      for (int r = 0; r < 8; ++r) {
        const int t = tbase + r + 8 * half;  // D: vgpr -> t row
        if (t < T_)
          mag[((size_t)b * F_ + f) * T_ + t] =
              sqrtf(acc_re[r] * acc_re[r] + acc_im[r] * acc_im[r]);
      }
    }
  }
}

// ---------------------------------------------------------------------------
// Kernel 2: time-pool + gate MLP + top-2 softmax routing. One block per batch.
// ---------------------------------------------------------------------------
__global__ __launch_bounds__(128) void gate_kernel(const float* __restrict__ mag,
    const float* __restrict__ gw1, const float* __restrict__ gb1,
    const float* __restrict__ gw2, const float* __restrict__ gb2,
    int* __restrict__ gidx, float* __restrict__ gwt) {
  const int b = blockIdx.x;
  const int tid = threadIdx.x;
  __shared__ float s_pooled[F_];
  __shared__ float s_h[H_];
  __shared__ float s_logits[E_];

  for (int f = tid; f < F_; f += 128) {
    const float* row = mag + ((size_t)b * F_ + f) * T_;
    float s = 0.0f;
    for (int t = 0; t < T_; ++t) s += row[t];
    s_pooled[f] = s / (float)T_;
  }
  __syncthreads();
  {
    float acc = gb1[tid];
    for (int f = 0; f < F_; ++f) acc += s_pooled[f] * gw1[f * H_ + tid];
    s_h[tid] = fmaxf(acc, 0.0f);
  }
  __syncthreads();
  if (tid < E_) {
    float acc = gb2[tid];
    for (int j = 0; j < H_; ++j) acc += s_h[j] * gw2[j * E_ + tid];
    s_logits[tid] = acc;
  }
  __syncthreads();
  if (tid == 0) {
    int i0 = 0; float v0 = s_logits[0];
    for (int e = 1; e < E_; ++e)
      if (s_logits[e] > v0) { v0 = s_logits[e]; i0 = e; }
    int i1 = -1; float v1 = -__builtin_inff();
    for (int e = 0; e < E_; ++e)
      if (e != i0 && s_logits[e] > v1) { v1 = s_logits[e]; i1 = e; }
    const float ex = expf(v1 - v0);          // stable 2-way softmax
    gidx[b * 2 + 0] = i0;  gidx[b * 2 + 1] = i1;
    gwt[b * 2 + 0] = 1.0f / (1.0f + ex);
    gwt[b * 2 + 1] = ex / (1.0f + ex);
  }
}

// ---------------------------------------------------------------------------
// Kernel 3: conv1 (129ch -> 64ch, k=5, same pad) + bias + relu + maxpool2.
// Block = 4 waves; weight tile (16 c_out x 132 c_pad x 5 taps) staged to LDS
// transposed to [tap][c][cout_local] (conflict-free B-fragment reads). Each
// wave holds 4 t-tile accumulators so each B fragment feeds 4 WMMAs. A loads
// are unconditional (clamped addresses) + post-select zero.
// grid (4 n-tiles, B*2 slots), block 128. Output t spans [0,256): t=256 never
// survives the VALID pool so 16 t-tiles suffice.
// ---------------------------------------------------------------------------
__global__ __launch_bounds__(128) void conv1_kernel(const float* __restrict__ mag,
    const float* __restrict__ w, const float* __restrict__ bias,
    const int* __restrict__ gidx, float* __restrict__ y1) {
  __shared__ float s_w[5 * 132 * 16];           // 42.2 KB
  const int nb  = blockIdx.x * 16;
  const int bs  = blockIdx.y, b = bs >> 1, slot = bs & 1;
  const int e   = gidx[b * 2 + slot];
  const int tid = threadIdx.x;
  const float* magb = mag + (size_t)b * F_ * T_;
  const float* we   = w + (size_t)e * C_ * F_ * 5;

  if (tid < F_) __builtin_prefetch(magb + (size_t)tid * T_, 0, 0);

  for (int i = tid; i < 5 * 132 * 16; i += 128) {
    const int cl = i & 15, kk = i >> 4;         // kk = tap*132 + c
    const int tap = kk / 132, c = kk - tap * 132;
    s_w[i] = (c < F_) ? we[((size_t)(nb + cl) * F_ + c) * 5 + tap] : 0.0f;
  }
  __syncthreads();

  const int lane = tid & 31, wave = tid >> 5;
  const int half = lane >> 4, l15 = lane & 15;
  const int cout = nb + l15;

  v8f acc[4] = {{}, {}, {}, {}};
  for (int tap = 0; tap < 5; ++tap) {
    for (int c0 = 0; c0 < 132; c0 += 4) {
      v2f bv;
      #pragma unroll
      for (int j = 0; j < 2; ++j)
        bv[j] = s_w[((tap * 132 + c0 + 2 * half + j) << 4) + l15];
      #pragma unroll
      for (int m = 0; m < 4; ++m) {
        const int trow = (wave * 4 + m) * 16 + l15 + tap - 2;
        const int trc  = min(max(trow, 0), T_ - 1);
        const bool tok = (trow >= 0) && (trow < T_);
        v2f av;
        #pragma unroll
        for (int j = 0; j < 2; ++j) {
          const int c  = c0 + 2 * half + j;
          const int cc = min(c, F_ - 1);
          const float v = magb[(size_t)cc * T_ + trc];
          av[j] = (tok && c < F_) ? v : 0.0f;
        }
        acc[m] = wmma4(av, bv, acc[m]);
      }
    }
  }
  const float bval = bias[e * C_ + cout];
  float* yb = y1 + (((size_t)b * 2 + slot) * C_ + cout) * T1_;
  #pragma unroll
  for (int m = 0; m < 4; ++m) {
    const int tbase = (wave * 4 + m) * 16;
    #pragma unroll
    for (int i = 0; i < 4; ++i) {               // in-lane maxpool2
      const float a0 = fmaxf(acc[m][2 * i]     + bval, 0.0f);
      const float a1 = fmaxf(acc[m][2 * i + 1] + bval, 0.0f);
      yb[(tbase >> 1) + 4 * half + i] = fmaxf(a0, a1);
    }
  }
}

// ---------------------------------------------------------------------------
// Kernel 4: conv2 (64->64, k=3) + bias + relu + maxpool2.
// Weight tile (16x64x3 = 12 KB) staged to LDS with GLOBAL_LOAD_ASYNC_TO_LDS
// (per-lane LDS destinations let us transpose during the async copy).
// grid (4 n-tiles, B*2), block 128 (4 waves x 2 t-tiles).
// ---------------------------------------------------------------------------
__global__ __launch_bounds__(128) void conv2_kernel(const float* __restrict__ y1,
    const float* __restrict__ w, const float* __restrict__ bias,
    const int* __restrict__ gidx, float* __restrict__ y2) {
  __shared__ float s_w[3 * C_ * 16];            // 12 KB, [tap][c][cout_local]
  const int nb  = blockIdx.x * 16;
  const int bs  = blockIdx.y, b = bs >> 1, slot = bs & 1;
  const int e   = gidx[b * 2 + slot];
  const int tid = threadIdx.x;
  const float* yin = y1 + (((size_t)b * 2 + slot) * C_) * T1_;
  const float* we  = w + ((size_t)e * C_ + nb) * C_ * 3;   // 16 couts, contiguous

  const uint32_t lds0 = lds_off(s_w);
  for (int i = tid; i < 16 * C_ * 3; i += 128) {           // 24 iters, uniform
    const int cl = i / 192, rem = i - cl * 192;
    const int c  = rem / 3,  tap = rem - c * 3;
    const uint32_t dst  = lds0 + (uint32_t)((((tap * C_ + c) << 4) + cl) << 2);
    const uint32_t voff = (uint32_t)(i << 2);
    asm volatile("global_load_async_to_lds_b32 %0, %1, %2"
                 :: "v"(dst), "v"(voff), "s"(we) : "memory");
  }
  asm volatile("s_wait_asynccnt 0x0" ::: "memory");
  __syncthreads();

  const int lane = tid & 31, wave = tid >> 5;
  const int half = lane >> 4, l15 = lane & 15;
  const int cout = nb + l15;

  v8f acc[2] = {{}, {}};
  for (int tap = 0; tap < 3; ++tap) {
    #pragma unroll 2
    for (int c0 = 0; c0 < C_; c0 += 4) {
      v2f bv;
      #pragma unroll
      for (int j = 0; j < 2; ++j)
        bv[j] = s_w[((tap * C_ + c0 + 2 * half + j) << 4) + l15];
      #pragma unroll
      for (int m = 0; m < 2; ++m) {
        const int trow = (wave * 2 + m) * 16 + l15 + tap - 1;
        const int trc  = min(max(trow, 0), T1_ - 1);
        const bool tok = (trow >= 0) && (trow < T1_);
        v2f av;
        #pragma unroll
        for (int j = 0; j < 2; ++j) {
          const int c = c0 + 2 * half + j;
          const float v = yin[(size_t)c * T1_ + trc];
          av[j] = tok ? v : 0.0f;
        }
        acc[m] = wmma4(av, bv, acc[m]);
      }
    }
  }
  const float bval = bias[e * C_ + cout];
  float* yb = y2 + (((size_t)b * 2 + slot) * C_ + cout) * T2_;
  #pragma unroll
  for (int m = 0; m < 2; ++m) {
    const int tbase = (wave * 2 + m) * 16;
    #pragma unroll
    for (int i = 0; i < 4; ++i) {
      const float a0 = fmaxf(acc[m][2 * i]     + bval, 0.0f);
      const float a1 = fmaxf(acc[m][2 * i + 1] + bval, 0.0f);
      yb[(tbase >> 1) + 4 * half + i] = fmaxf(a0, a1);
    }
  }
}

// ---------------------------------------------------------------------------
// Kernel 5: conv3 (64->64, k=3, no pool) for both routed slots + gated combine.
// Both slots' weight tiles (2 x 12 KB) staged via async-to-LDS. grid
// (4 n-tiles, B), block 128 (4 waves x 1 t-tile, 2 slot accumulators each).
// Every output element is written -> no pre-zeroing of d_out needed.
// ---------------------------------------------------------------------------
__global__ __launch_bounds__(128) void conv3_combine_kernel(const float* __restrict__ y2,
    const float* __restrict__ w, const float* __restrict__ bias,
    const int* __restrict__ gidx, const float* __restrict__ gwt,
    float* __restrict__ out) {
  __shared__ float s_w[2 * 3 * C_ * 16];        // 24 KB, [slot][tap][c][cout_l]
  const int nb  = blockIdx.x * 16;
  const int b   = blockIdx.y;
  const int tid = threadIdx.x;
  const int e0  = gidx[b * 2 + 0], e1 = gidx[b * 2 + 1];
  const float* we0 = w + ((size_t)e0 * C_ + nb) * C_ * 3;
  const float* we1 = w + ((size_t)e1 * C_ + nb) * C_ * 3;

  const uint32_t lds0 = lds_off(s_w);
  for (int i = tid; i < 16 * C_ * 3; i += 128) {
    const int cl = i / 192, rem = i - cl * 192;
    const int c  = rem / 3,  tap = rem - c * 3;
    const uint32_t off  = (uint32_t)((((tap * C_ + c) << 4) + cl) << 2);
    const uint32_t voff = (uint32_t)(i << 2);
    asm volatile("global_load_async_to_lds_b32 %0, %1, %2"
                 :: "v"(lds0 + off), "v"(voff), "s"(we0) : "memory");
    asm volatile("global_load_async_to_lds_b32 %0, %1, %2"
                 :: "v"(lds0 + 12288u + off), "v"(voff), "s"(we1) : "memory");
  }
  asm volatile("s_wait_asynccnt 0x0" ::: "memory");
  __syncthreads();

  const int lane = tid & 31, wave = tid >> 5;
  const int half = lane >> 4, l15 = lane & 15;
  const int cout  = nb + l15;
  const int tbase = wave * 16;

  v8f acc[2] = {{}, {}};
  for (int tap = 0; tap < 3; ++tap) {
    const int trow = tbase + l15 + tap - 1;
    const int trc  = min(max(trow, 0), T2_ - 1);
    const bool tok = (trow >= 0) && (trow < T2_);
    #pragma unroll 2
    for (int c0 = 0; c0 < C_; c0 += 4) {
      #pragma unroll
      for (int slot = 0; slot < 2; ++slot) {
        const float* yin = y2 + (((size_t)b * 2 + slot) * C_) * T2_;
        v2f av, bv;
        #pragma unroll
        for (int j = 0; j < 2; ++j) {
          const int c = c0 + 2 * half + j;
          const float v = yin[(size_t)c * T2_ + trc];
          av[j] = tok ? v : 0.0f;
          bv[j] = s_w[slot * 3072 + ((tap * C_ + c) << 4) + l15];
        }
        acc[slot] = wmma4(av, bv, acc[slot]);
      }
    }
  }
  const float w0 = gwt[b * 2 + 0], w1 = gwt[b * 2 + 1];
  const float b0 = bias[e0 * C_ + cout], b1 = bias[e1 * C_ + cout];
  #pragma unroll
  for (int r = 0; r < 8; ++r) {
    const int t = tbase + r + 8 * half;
    out[((size_t)b * C_ + cout) * T2_ + t] =
        w0 * fmaxf(acc[0][r] + b0, 0.0f) + w1 * fmaxf(acc[1][r] + b1, 0.0f);
  }
}

extern "C" void kernel_launch(void* const* d_in, const int* in_sizes, int n_in,
                              void* d_out, int out_size, void* d_ws, size_t ws_size,
                              hipStream_t stream) {
  const float* x   = (const float*)d_in[0];
  const float* gw1 = (const float*)d_in[1];
  const float* gb1 = (const float*)d_in[2];
  const float* gw2 = (const float*)d_in[3];
  const float* gb2 = (const float*)d_in[4];
  const float* c1w = (const float*)d_in[5];
  const float* c1b = (const float*)d_in[6];
  const float* c2w = (const float*)d_in[7];
  const float* c2b = (const float*)d_in[8];
  const float* c3w = (const float*)d_in[9];
  const float* c3b = (const float*)d_in[10];
  float* out = (float*)d_out;

  // Workspace carve-up (~30 MB of fp32 intermediates, L2 resident).
  float* mag  = (float*)d_ws;                       // B*F*T
  int*   gidx = (int*)(mag + (size_t)B_ * F_ * T_); // B*2 routed expert ids
  float* gwt  = (float*)(gidx + B_ * 2);            // B*2 gate weights
  float* y1   = gwt + B_ * 2;                       // B*2*C*T1
  float* y2   = y1 + (size_t)B_ * 2 * C_ * T1_;     // B*2*C*T2

  stft_mag_kernel<<<dim3(9, B_), 128, 0, stream>>>(x, mag);
  gate_kernel<<<dim3(B_), 128, 0, stream>>>(mag, gw1, gb1, gw2, gb2, gidx, gwt);
  conv1_kernel<<<dim3(4, B_ * 2), 128, 0, stream>>>(mag, c1w, c1b, gidx, y1);
  conv2_kernel<<<dim3(4, B_ * 2), 128, 0, stream>>>(y1, c2w, c2b, gidx, y2);
  conv3_combine_kernel<<<dim3(4, B_), 128, 0, stream>>>(y2, c3w, c3b, gidx, gwt, out);
}